// YoloLoss_48954037239810
// MI455X (gfx1250) — compile-verified
//
#include <hip/hip_runtime.h>

typedef float v2f __attribute__((ext_vector_type(2)));
typedef float v4f __attribute__((ext_vector_type(4)));
typedef float v8f __attribute__((ext_vector_type(8)));

#define YS      7
#define YBATCH  32768
#define YCELLS  (YBATCH * YS * YS)   /* 1,605,632 */
#define YGROUPS (YCELLS / 4)         /* 401,408 : 4 cells per thread */
#define YBLOCK  256
#define YGRID   (YGROUPS / YBLOCK)   /* 1568, exact */

// ---------------------------------------------------------------------------
// IOU exactly per reference (EPS=1e-6, clip at 0, union may be negative)
// ---------------------------------------------------------------------------
__device__ __forceinline__ float iou_f(float ax, float ay, float aw, float ah,
                                       float bx, float by, float bw, float bh) {
    float ax1 = ax - aw * 0.5f, ay1 = ay - ah * 0.5f;
    float ax2 = ax + aw * 0.5f, ay2 = ay + ah * 0.5f;
    float bx1 = bx - bw * 0.5f, by1 = by - bh * 0.5f;
    float bx2 = bx + bw * 0.5f, by2 = by + bh * 0.5f;
    float iw = fmaxf(fminf(ax2, bx2) - fmaxf(ax1, bx1), 0.0f);
    float ih = fmaxf(fminf(ay2, by2) - fmaxf(ay1, by1), 0.0f);
    float inter = iw * ih;
    float uni = (ax2 - ax1) * (ay2 - ay1) + (bx2 - bx1) * (by2 - by1) - inter;
    return inter / (uni + 1e-6f);
}

// ---------------------------------------------------------------------------
// Exact f32 wave(32) sum via V_WMMA_F32_16X16X4_F32 with a ones B-matrix.
// A layout (16x4 f32): lane m      : A[m,0]=acc (v0), A[m,1]=0 (v1)
//                      lane m+16   : A[m,2]=acc (v0), A[m,3]=0 (v1)
// => D[m,n] = acc[m] + acc[m+16]  for all n.
// D VGPR j holds rows j (lanes 0-15) and j+8 (lanes 16-31), so
// t = sum_j d[j] gives rows 0..7 summed on lanes 0-15 and rows 8..15 on
// lanes 16-31; one xor-16 shuffle completes the wave total (all lanes).
// Requires EXEC == all ones (guaranteed: full blocks, no early exits).
// ---------------------------------------------------------------------------
__device__ __forceinline__ float wave_sum_wmma(float acc) {
    v2f a; a.x = acc;  a.y = 0.0f;
    v2f b; b.x = 1.0f; b.y = 1.0f;
    v8f c = {};
    v8f d = __builtin_amdgcn_wmma_f32_16x16x4_f32(
        /*neg_a=*/false, a, /*neg_b=*/false, b,
        /*c_mod=*/(short)0, c, /*reuse_a=*/false, /*reuse_b=*/false);
    float t = d[0] + d[1] + d[2] + d[3] + d[4] + d[5] + d[6] + d[7];
    t += __shfl_xor(t, 16, 32);
    return t;
}

// ---------------------------------------------------------------------------
// Stage 1: each thread consumes 4 consecutive cells (16B-aligned b128 loads),
// block-reduces deterministically, writes one partial per block.
// ---------------------------------------------------------------------------
__global__ __launch_bounds__(YBLOCK) void yolo_partial_kernel(
        const float* __restrict__ pred,   // [B,7,7,13]
        const float* __restrict__ tgt,    // [B,7,7,8]
        float* __restrict__ partial) {    // [YGRID]
    const size_t g = (size_t)blockIdx.x * YBLOCK + threadIdx.x;   // group of 4 cells
    const v4f* p4 = (const v4f*)(pred + g * 52);                  // 208 B, 16B aligned
    const v4f* t4 = (const v4f*)(tgt  + g * 32);                  // 128 B, 16B aligned

    float pp[52];
    float tt[32];
#pragma unroll
    for (int i = 0; i < 13; ++i) {
        v4f v = p4[i];
        pp[4*i+0] = v.x; pp[4*i+1] = v.y; pp[4*i+2] = v.z; pp[4*i+3] = v.w;
    }
#pragma unroll
    for (int i = 0; i < 8; ++i) {
        v4f v = t4[i];
        tt[4*i+0] = v.x; tt[4*i+1] = v.y; tt[4*i+2] = v.z; tt[4*i+3] = v.w;
    }

    float acc = 0.0f;
#define PF(k) pp[13 * cc + (k)]
#define TF(k) tt[ 8 * cc + (k)]
#pragma unroll
    for (int cc = 0; cc < 4; ++cc) {
        const float m   = (TF(0) == 1.0f) ? 1.0f : 0.0f;
        const float tx = TF(1), ty = TF(2), tw = TF(3), th = TF(4);

        const float iou1 = iou_f(PF(1), PF(2), PF(3), PF(4), tx, ty, tw, th);
        const float iou2 = iou_f(PF(6), PF(7), PF(8), PF(9), tx, ty, tw, th);
        const bool  pick1 = iou1 > iou2;
        const float iou = pick1 ? iou1 : iou2;

        const float sc = pick1 ? PF(0) : PF(5);   // selected confidence
        const float sx = pick1 ? PF(1) : PF(6);
        const float sy = pick1 ? PF(2) : PF(7);
        const float sw = pick1 ? PF(3) : PF(8);
        const float sh = pick1 ? PF(4) : PF(9);
        const float uc = pick1 ? PF(5) : PF(0);   // unselected confidence

        const float dx = sx - tx, dy = sy - ty;
        const float loss_xy = dx * dx + dy * dy;

        const float wt = copysignf(sqrtf(fabsf(sw)), sw) - sqrtf(tw);
        const float ht = copysignf(sqrtf(fabsf(sh)), sh) - sqrtf(th);
        const float loss_wh = wt * wt + ht * ht;

        const float dob = sc - iou;
        const float loss_obj = dob * dob;

        const float c0 = PF(10) - TF(5);
        const float c1 = PF(11) - TF(6);
        const float c2 = PF(12) - TF(7);
        const float loss_cls = c0 * c0 + c1 * c1 + c2 * c2;

        const float noobj_both = PF(0) * PF(0) + PF(5) * PF(5);
        const float loss_noobj = m * uc * uc + (1.0f - m) * noobj_both;

        acc += m * (5.0f * (loss_xy + loss_wh) + loss_obj + loss_cls)
             + 0.5f * loss_noobj;
    }
#undef PF
#undef TF

    const float wsum = wave_sum_wmma(acc);

    __shared__ float warp_sums[YBLOCK / 32];
    const int lane = threadIdx.x & 31;
    const int wid  = threadIdx.x >> 5;
    if (lane == 0) warp_sums[wid] = wsum;
    __syncthreads();
    if (threadIdx.x == 0) {
        float s = 0.0f;
#pragma unroll
        for (int i = 0; i < YBLOCK / 32; ++i) s += warp_sums[i];  // fixed order
        partial[blockIdx.x] = s;
    }
}

// ---------------------------------------------------------------------------
// Stage 2: single block deterministically folds the 1568 block partials.
// ---------------------------------------------------------------------------
__global__ __launch_bounds__(YBLOCK) void yolo_final_kernel(
        const float* __restrict__ partial, float* __restrict__ out) {
    float acc = 0.0f;
    for (int i = threadIdx.x; i < YGRID; i += YBLOCK) acc += partial[i];

    const float wsum = wave_sum_wmma(acc);   // loop reconverged; EXEC all ones

    __shared__ float warp_sums[YBLOCK / 32];
    const int lane = threadIdx.x & 31;
    const int wid  = threadIdx.x >> 5;
    if (lane == 0) warp_sums[wid] = wsum;
    __syncthreads();
    if (threadIdx.x == 0) {
        float s = 0.0f;
#pragma unroll
        for (int i = 0; i < YBLOCK / 32; ++i) s += warp_sums[i];
        out[0] = s * (1.0f / (float)YBATCH);
    }
}

extern "C" void kernel_launch(void* const* d_in, const int* in_sizes, int n_in,
                              void* d_out, int out_size, void* d_ws, size_t ws_size,
                              hipStream_t stream) {
    const float* pred = (const float*)d_in[0];   // predictions f32 [32768,7,7,13]
    const float* tgt  = (const float*)d_in[1];   // target      f32 [32768,7,7,8]
    float* partial = (float*)d_ws;               // YGRID floats of scratch
    float* out = (float*)d_out;

    yolo_partial_kernel<<<YGRID, YBLOCK, 0, stream>>>(pred, tgt, partial);
    yolo_final_kernel<<<1, YBLOCK, 0, stream>>>(partial, out);
}